// Grid_30743375905190
// MI455X (gfx1250) — compile-verified
//
#include <hip/hip_runtime.h>
#include <hip/hip_bf16.h>

// 3D trilinear grid-sample, MI455X-tuned:
//  - 64MB volume stays resident in the 192MB L2 (gathers use default RT hint)
//  - coords / output are streamed with non-temporal hints so they never evict
//    volume lines (TH=NT on global_load/global_store)
//  - all 8 gathers issued before any math -> one s_wait_loadcnt, deep MLP
//  - wave32, tiny VGPR footprint -> max waves in flight to hide L2 latency
//  - global_prefetch_b8 on the coords stream a few KB ahead

#define GDIM 256

__global__ __launch_bounds__(256)
void grid_sample3d_kernel(const float* __restrict__ coords,
                          const float* __restrict__ vol,
                          float* __restrict__ out,
                          int P) {
    int i = blockIdx.x * blockDim.x + threadIdx.x;
    if (i >= P) return;

    const float* cp = coords + 3 * (size_t)i;

    // Prefetch the coords stream ~12KB ahead (speculative; OOB is silently
    // dropped per CDNA5 prefetch semantics). Lowers to global_prefetch_b8.
    __builtin_prefetch(cp + 3 * 1024, 0, 1);

    // Streaming loads: non-temporal so they do not displace the L2-resident
    // volume working set.
    float cx = __builtin_nontemporal_load(cp + 0);
    float cy = __builtin_nontemporal_load(cp + 1);
    float cz = __builtin_nontemporal_load(cp + 2);

    const float S = 0.5f * (float)(GDIM - 1);   // (x+1)/2 * (G-1)
    float ix = (cx + 1.0f) * S;
    float iy = (cy + 1.0f) * S;
    float iz = (cz + 1.0f) * S;

    float xf = floorf(ix), yf = floorf(iy), zf = floorf(iz);
    float fx = ix - xf,   fy = iy - yf,   fz = iz - zf;

    int x0 = (int)xf, y0 = (int)yf, z0 = (int)zf;
    // Clamp exactly as the reference does (coords in [-1,1] -> idx in [0,255],
    // but guard the boundary where floor hits G-1).
    x0 = max(0, min(x0, GDIM - 1));
    y0 = max(0, min(y0, GDIM - 1));
    z0 = max(0, min(z0, GDIM - 1));
    int x1 = min(x0 + 1, GDIM - 1);
    int y1 = min(y0 + 1, GDIM - 1);
    int z1 = min(z0 + 1, GDIM - 1);

    // Flat index: z*65536 + y*256 + x  (max 16,777,215 -> i32 offsets, so the
    // gathers use SADDR base + 32-bit VGPR offset addressing).
    int zy00 = (z0 << 16) + (y0 << 8);
    int zy01 = (z0 << 16) + (y1 << 8);
    int zy10 = (z1 << 16) + (y0 << 8);
    int zy11 = (z1 << 16) + (y1 << 8);

    // Issue all 8 gathers up front (default RT temporal hint: volume lines
    // stay hot in L2). Compiler batches these under one s_wait_loadcnt.
    float v000 = vol[zy00 + x0];
    float v001 = vol[zy00 + x1];
    float v010 = vol[zy01 + x0];
    float v011 = vol[zy01 + x1];
    float v100 = vol[zy10 + x0];
    float v101 = vol[zy10 + x1];
    float v110 = vol[zy11 + x0];
    float v111 = vol[zy11 + x1];

    // Factored trilinear: 7 lerps (fmaf) instead of 8 weight products.
    float c00 = fmaf(fx, v001 - v000, v000);
    float c01 = fmaf(fx, v011 - v010, v010);
    float c10 = fmaf(fx, v101 - v100, v100);
    float c11 = fmaf(fx, v111 - v110, v110);

    float c0 = fmaf(fy, c01 - c00, c00);
    float c1 = fmaf(fy, c11 - c10, c10);

    float r = fmaf(fz, c1 - c0, c0);

    // Streaming store: non-temporal, don't pollute L2.
    __builtin_nontemporal_store(r, out + i);
}

extern "C" void kernel_launch(void* const* d_in, const int* in_sizes, int n_in,
                              void* d_out, int out_size, void* d_ws, size_t ws_size,
                              hipStream_t stream) {
    const float* coords = (const float*)d_in[0];   // (P, 3) float32
    const float* grid   = (const float*)d_in[1];   // (1,1,256,256,256) float32
    float* out = (float*)d_out;                    // (P, 1) float32

    int P = in_sizes[0] / 3;
    const int block = 256;                          // 8 wave32 waves / block
    int nblocks = (P + block - 1) / block;

    grid_sample3d_kernel<<<nblocks, block, 0, stream>>>(coords, grid, out, P);
}